// GraphSAGE_79328045957723
// MI455X (gfx1250) — compile-verified
//
#include <hip/hip_runtime.h>

#define NN 100000
#define NE 1600000
#define DD 64

typedef float v2f __attribute__((ext_vector_type(2)));
typedef float v8f __attribute__((ext_vector_type(8)));

// ---------------------------------------------------------------------------
// Zero a range of 32-bit words (used for deg + agg buffers each call).
// ---------------------------------------------------------------------------
__global__ void zero_words(unsigned int* __restrict__ p, long long n) {
    long long i = (long long)blockIdx.x * blockDim.x + threadIdx.x;
    long long stride = (long long)gridDim.x * blockDim.x;
    for (; i < n; i += stride) p[i] = 0u;
}

// ---------------------------------------------------------------------------
// In-degree: one thread per edge.
// ---------------------------------------------------------------------------
__global__ void edge_degree(const int* __restrict__ dst, int* __restrict__ deg) {
    int e = blockIdx.x * blockDim.x + threadIdx.x;
    if (e < NE) atomicAdd(&deg[dst[e]], 1);
}

// ---------------------------------------------------------------------------
// Edge scatter-add: agg[dst] += h[src]. One thread per (edge, 4-float group).
// float4 gather of the source row slice, 4 global fp32 atomics to the dest.
// This is the bandwidth-dominant phase (~410 MB random gather per layer).
// ---------------------------------------------------------------------------
__global__ __launch_bounds__(256)
void edge_scatter(const float* __restrict__ h,
                  const int* __restrict__ src,
                  const int* __restrict__ dst,
                  float* __restrict__ agg) {
    long long tid = (long long)blockIdx.x * blockDim.x + threadIdx.x;
    int e = (int)(tid >> 4);
    int g = (int)(tid & 15);
    if (e >= NE) return;
    int s = src[e];
    int d = dst[e];
    const float* hp = h + (size_t)s * DD + g * 4;
    __builtin_prefetch(hp, 0, 1);                  // global_prefetch_b8
    float4 v = *(const float4*)hp;
    float* a = agg + (size_t)d * DD + g * 4;
    atomicAdd(a + 0, v.x);
    atomicAdd(a + 1, v.y);
    atomicAdd(a + 2, v.z);
    atomicAdd(a + 3, v.w);
}

// ---------------------------------------------------------------------------
// Node update via fp32 WMMA:
//   out[16xDD tile] = hs @ Wself + (agg/max(deg,1)) @ Wneigh + bias
// One wave handles 16 node rows and all 64 output columns.
// K=64 consumed in 16 steps of V_WMMA_F32_16X16X4_F32 per weight matrix.
//
// Both 64x64 weight matrices are staged ONCE per block into LDS, pre-swizzled
// into the WMMA B-fragment layout so the inner loop does a single conflict-
// free ds_load_b64 per (tile, matrix) instead of 8 strided global loads.
//
// fp32 WMMA operand layouts (CDNA5 ISA 7.12.2):
//   A (16x4, MxK): lane l -> row m=l&15, kHalf=(l>>4); a[j]=A[m][2*kHalf+j]
//   B (4x16, KxN): lane l -> col n=l&15, kHalf=(l>>4); b[j]=B[2*kHalf+j][n]
//   C/D (16x16):   lane l -> col n=l&15; vgpr i -> row 8*(l>>4)+i
// ---------------------------------------------------------------------------
__global__ __launch_bounds__(256)
void sage_node_wmma(const float* __restrict__ hs,      // self features [N,64]
                    const float* __restrict__ agg,     // summed neighbors [N,64]
                    const int* __restrict__ deg,       // in-degree [N]
                    const float* __restrict__ Wself,   // [64,64] row-major (k,n)
                    const float* __restrict__ Wneigh,  // [64,64] row-major (k,n)
                    const float* __restrict__ bias,    // [64]
                    float* __restrict__ out,           // [N,64]
                    int n_nodes) {
    // B fragments: frag f = (kstep*4 + tile) in 0..63, 32 lanes, 2 floats each.
    __shared__ v2f wl[2][64 * 32];                     // 32 KB total

    const int lane = threadIdx.x & 31;

    // ---- cooperative LDS fill, pre-swizzled to B-fragment layout ----
    for (int idx = threadIdx.x; idx < 64 * 32; idx += blockDim.x) {
        const int f  = idx >> 5;           // fragment 0..63
        const int l  = idx & 31;           // lane slot
        const int kk = f >> 2;             // k-step 0..15
        const int t  = f & 3;              // column tile 0..3
        const int kA = kk * 4 + (l >> 4) * 2;
        const int nc = t * 16 + (l & 15);
        v2f s, n;
        s[0] = Wself [(kA + 0) * DD + nc];
        s[1] = Wself [(kA + 1) * DD + nc];
        n[0] = Wneigh[(kA + 0) * DD + nc];
        n[1] = Wneigh[(kA + 1) * DD + nc];
        wl[0][idx] = s;
        wl[1][idx] = n;
    }
    __syncthreads();

    const int wave = blockIdx.x * (blockDim.x >> 5) + (threadIdx.x >> 5);
    const int row0 = wave * 16;
    if (row0 < n_nodes) {                  // wave-uniform: EXEC all-1s inside
        const int mLoc  = lane & 15;
        const int kHalf = lane >> 4;       // 0 or 1
        const int m     = row0 + mLoc;
        const int mc    = (m < n_nodes) ? m : (n_nodes - 1);  // clamp loads

        // mean normalization factor for this row
        const float inv = 1.0f / fmaxf((float)deg[mc], 1.0f);

        v8f acc[4] = {};                   // 4 column tiles of 16x16 f32

        #pragma unroll 4
        for (int k = 0; k < DD; k += 4) {
            const int kA = k + kHalf * 2;
            // A operands: self row slice and normalized neighbor row slice
            v2f aS = *(const v2f*)(hs  + (size_t)mc * DD + kA);
            v2f tN = *(const v2f*)(agg + (size_t)mc * DD + kA);
            v2f aN;
            aN[0] = tN[0] * inv;
            aN[1] = tN[1] * inv;
            const int fBase = k * 32;      // (k>>2)*4 frags * 32 lanes
            #pragma unroll
            for (int t = 0; t < 4; ++t) {
                v2f bS = wl[0][fBase + t * 32 + lane];   // ds_load_b64
                v2f bN = wl[1][fBase + t * 32 + lane];   // conflict-free
                acc[t] = __builtin_amdgcn_wmma_f32_16x16x4_f32(
                    false, aS, false, bS, (short)0, acc[t], false, false);
                acc[t] = __builtin_amdgcn_wmma_f32_16x16x4_f32(
                    false, aN, false, bN, (short)0, acc[t], false, false);
            }
        }

        // store D tiles + bias
        const int nLane = lane & 15;
        #pragma unroll
        for (int t = 0; t < 4; ++t) {
            const int ncol = t * 16 + nLane;
            const float bv = bias[ncol];
            #pragma unroll
            for (int i = 0; i < 8; ++i) {
                const int r = row0 + kHalf * 8 + i;
                if (r < n_nodes) out[(size_t)r * DD + ncol] = acc[t][i] + bv;
            }
        }
    }
}

// ---------------------------------------------------------------------------
// Host-side orchestration (graph-capture safe: kernels only, all on stream).
// ---------------------------------------------------------------------------
extern "C" void kernel_launch(void* const* d_in, const int* in_sizes, int n_in,
                              void* d_out, int out_size, void* d_ws, size_t ws_size,
                              hipStream_t stream) {
    const float* x       = (const float*)d_in[0];
    const int*   src     = (const int*)d_in[1];
    const int*   dst     = (const int*)d_in[2];
    const float* Wself1  = (const float*)d_in[3];
    const float* Wneigh1 = (const float*)d_in[4];
    const float* b1      = (const float*)d_in[5];
    const float* Wself2  = (const float*)d_in[6];
    const float* Wneigh2 = (const float*)d_in[7];
    const float* b2      = (const float*)d_in[8];
    float* out = (float*)d_out;

    // workspace layout: [deg: N ints][h1: N*64 f32][agg: N*64 f32]
    char* ws = (char*)d_ws;
    size_t degBytes = ((size_t)NN * sizeof(int) + 255) & ~(size_t)255;
    int*   deg = (int*)ws;
    float* h1  = (float*)(ws + degBytes);
    float* agg = h1 + (size_t)NN * DD;

    const int TB = 256;
    const long long aggWords = (long long)NN * DD;
    const long long degWords = NN;
    const int zeroBlocks = 2048;
    const int edgeBlocks = (NE + TB - 1) / TB;                       // one thread/edge
    const int scatBlocks = (int)(((long long)NE * 16 + TB - 1) / TB);// edge x 16 groups
    const int waves      = (NN + 15) / 16;
    const int nodeBlocks = (waves + (TB / 32) - 1) / (TB / 32);

    // ---- degrees (shared by both layers) ----
    zero_words<<<zeroBlocks, TB, 0, stream>>>((unsigned int*)deg, degWords);
    edge_degree<<<edgeBlocks, TB, 0, stream>>>(dst, deg);

    // ---- layer 1: agg = scatter(x); h1 = x@Ws1 + (agg/deg)@Wn1 + b1 ----
    zero_words<<<zeroBlocks, TB, 0, stream>>>((unsigned int*)agg, aggWords);
    edge_scatter<<<scatBlocks, TB, 0, stream>>>(x, src, dst, agg);
    sage_node_wmma<<<nodeBlocks, TB, 0, stream>>>(x, agg, deg, Wself1, Wneigh1,
                                                  b1, h1, NN);

    // ---- layer 2: agg = scatter(h1); out = h1@Ws2 + (agg/deg)@Wn2 + b2 ----
    zero_words<<<zeroBlocks, TB, 0, stream>>>((unsigned int*)agg, aggWords);
    edge_scatter<<<scatBlocks, TB, 0, stream>>>(h1, src, dst, agg);
    sage_node_wmma<<<nodeBlocks, TB, 0, stream>>>(h1, agg, deg, Wself2, Wneigh2,
                                                  b2, out, NN);
}